// clusteringEMA_torch_73237782331475
// MI455X (gfx1250) — compile-verified
//
#include <hip/hip_runtime.h>
#include <hip/hip_bf16.h>

#define BATCH 65536
#define VEC   512
#define KCLUS 1024
#define GAMMA_ 0.99f
#define EPS_   1e-5f

#define ROWS_PER_BLOCK 128   // 4 waves x 32 rows (2 M-tiles per wave -> 2x B reuse)
#define XPAD 516             // LDS row stride in floats (512+4 kills 16-way bank conflicts)

typedef __attribute__((ext_vector_type(2))) float v2f;
typedef __attribute__((ext_vector_type(8))) float v8f;

// workspace layout (floats):
// [0, VEC*KCLUS)            embed_sum  (2 MB)
// [VEC*KCLUS, +KCLUS)       counts
// next KCLUS                w2
// next KCLUS                cs
#define ESUM_OFF   0
#define COUNTS_OFF (VEC * KCLUS)
#define W2_OFF     (COUNTS_OFF + KCLUS)
#define CS_OFF     (W2_OFF + KCLUS)

__global__ void zero_ws_kernel(float* __restrict__ ws, int n) {
    int i = blockIdx.x * blockDim.x + threadIdx.x;
    if (i < n) ws[i] = 0.0f;
}

__global__ void w2_kernel(const float* __restrict__ W, float* __restrict__ w2) {
    int k = blockIdx.x * blockDim.x + threadIdx.x;
    if (k >= KCLUS) return;
    float s = 0.0f;
#pragma unroll 8
    for (int v = 0; v < VEC; ++v) {
        float w = W[(size_t)v * KCLUS + k];   // consecutive lanes -> consecutive addrs
        s += w * w;
    }
    w2[k] = s;
}

__global__ __launch_bounds__(128)
void assign_accum_kernel(const float* __restrict__ X,
                         const float* __restrict__ W,
                         const float* __restrict__ w2,
                         float* __restrict__ esum,
                         float* __restrict__ counts)
{
    extern __shared__ char smem[];
    float* xs   = (float*)smem;                                         // 128 x 516 floats
    int*   aidx = (int*)(smem + ROWS_PER_BLOCK * XPAD * sizeof(float)); // 128 ints

    const int tid  = threadIdx.x;
    const int lane = tid & 31;
    const int wave = tid >> 5;
    const int rowBase = blockIdx.x * ROWS_PER_BLOCK;

    // ---- stage X tile (128 x 512) into LDS, float4-coalesced ----
    {
        const float4* Xg = (const float4*)(X + (size_t)rowBase * VEC);
#pragma unroll 4
        for (int i = tid; i < ROWS_PER_BLOCK * (VEC / 4); i += 128) {
            int row = i >> 7;       // /128 float4s per row
            int v4  = i & 127;
            float4 val = Xg[(size_t)row * (VEC / 4) + v4];
            *(float4*)&xs[row * XPAD + v4 * 4] = val;
        }
    }
    __syncthreads();

    // ---- WMMA f32 16x16x4: this wave owns TWO 16-row stripes (B-frag reused 2x) ----
    const int stripe0 = wave * 32;
    const int stripe1 = stripe0 + 16;
    const int l15     = lane & 15;
    const int khalf   = (lane >> 4) << 1;   // lanes 0-15: K=0,1 ; lanes 16-31: K=2,3

    float best0[8], best1[8];
    int   bidx0[8], bidx1[8];
#pragma unroll
    for (int r = 0; r < 8; ++r) {
        best0[r] = 3.4e38f; bidx0[r] = 0;
        best1[r] = 3.4e38f; bidx1[r] = 0;
    }

    const float* xrow0 = &xs[(stripe0 + l15) * XPAD + khalf];  // A-frag sources (reused 64x)
    const float* xrow1 = &xs[(stripe1 + l15) * XPAD + khalf];

    for (int kb = 0; kb < KCLUS / 16; ++kb) {
        const int col = kb * 16 + l15;                         // this lane's B/N column
        v8f acc0 = {0.f, 0.f, 0.f, 0.f, 0.f, 0.f, 0.f, 0.f};
        v8f acc1 = {0.f, 0.f, 0.f, 0.f, 0.f, 0.f, 0.f, 0.f};
        const float* wp = W + (size_t)khalf * KCLUS + col;
#pragma unroll 4
        for (int s = 0; s < VEC / 4; ++s) {
            v2f b;
            b.x = wp[0];                                       // W[(4s+kh)  ][col]
            b.y = wp[KCLUS];                                   // W[(4s+kh+1)][col]
            wp += 4 * KCLUS;
            v2f a0 = *(const v2f*)(xrow0 + 4 * s);             // ds_load_b64, bank-clean
            v2f a1 = *(const v2f*)(xrow1 + 4 * s);
            acc0 = __builtin_amdgcn_wmma_f32_16x16x4_f32(
                       false, a0, false, b, (short)0, acc0, false, false);
            acc1 = __builtin_amdgcn_wmma_f32_16x16x4_f32(
                       false, a1, false, b, (short)0, acc1, false, false);
        }
        const float w2c = w2[col];
#pragma unroll
        for (int r = 0; r < 8; ++r) {
            float sc0 = w2c - 2.0f * acc0[r];                  // argmin-equivalent score
            if (sc0 < best0[r]) { best0[r] = sc0; bidx0[r] = col; }
            float sc1 = w2c - 2.0f * acc1[r];
            if (sc1 < best1[r]) { best1[r] = sc1; bidx1[r] = col; }
        }
    }

    // ---- 16-lane min-reduce per row (xor masks stay within half-waves) ----
#pragma unroll
    for (int m = 8; m >= 1; m >>= 1) {
#pragma unroll
        for (int r = 0; r < 8; ++r) {
            float ob0 = __shfl_xor(best0[r], m, 32);
            int   oi0 = __shfl_xor(bidx0[r], m, 32);
            if (ob0 < best0[r] || (ob0 == best0[r] && oi0 < bidx0[r])) {
                best0[r] = ob0; bidx0[r] = oi0;
            }
            float ob1 = __shfl_xor(best1[r], m, 32);
            int   oi1 = __shfl_xor(bidx1[r], m, 32);
            if (ob1 < best1[r] || (ob1 == best1[r] && oi1 < bidx1[r])) {
                best1[r] = ob1; bidx1[r] = oi1;
            }
        }
    }
    if (l15 == 0) {
        int rb0 = stripe0 + (lane >> 4) * 8;  // lower half: rows 0-7, upper: rows 8-15
        int rb1 = stripe1 + (lane >> 4) * 8;
#pragma unroll
        for (int r = 0; r < 8; ++r) {
            aidx[rb0 + r] = bidx0[r];
            aidx[rb1 + r] = bidx1[r];
        }
    }
    __syncthreads();

    // ---- scatter: counts + embed_sum (L2-resident atomics), reusing LDS tile ----
    if (tid < ROWS_PER_BLOCK) atomicAdd(&counts[aidx[tid]], 1.0f);

    for (int i = tid; i < ROWS_PER_BLOCK * VEC; i += 128) {
        int row = i >> 9;          // /512
        int v   = i & (VEC - 1);
        int k   = aidx[row];
        atomicAdd(&esum[(size_t)v * KCLUS + k], xs[row * XPAD + v]);
    }
}

__global__ __launch_bounds__(1024)
void finalize1_kernel(const float* __restrict__ cluster_size,
                      const float* __restrict__ counts,
                      float* __restrict__ cs)
{
    __shared__ float red[KCLUS];
    int k = threadIdx.x;
    float c = counts[k];
    if (c == 0.0f) c = 1.0f;
    float ncs = GAMMA_ * cluster_size[k] + (1.0f - GAMMA_) * c;
    red[k] = ncs;
    __syncthreads();
    for (int s = KCLUS / 2; s > 0; s >>= 1) {
        if (k < s) red[k] += red[k + s];
        __syncthreads();
    }
    float n = red[0];
    cs[k] = (ncs + EPS_) / (n + (float)KCLUS * EPS_) * n;
}

__global__ void finalize2_kernel(const float* __restrict__ embed_avg,
                                 const float* __restrict__ esum,
                                 const float* __restrict__ cs,
                                 float* __restrict__ out)
{
    int i = blockIdx.x * blockDim.x + threadIdx.x;
    if (i >= VEC * KCLUS) return;
    int k = i & (KCLUS - 1);
    out[i] = (GAMMA_ * embed_avg[i] + (1.0f - GAMMA_) * esum[i]) / cs[k];
}

extern "C" void kernel_launch(void* const* d_in, const int* in_sizes, int n_in,
                              void* d_out, int out_size, void* d_ws, size_t ws_size,
                              hipStream_t stream) {
    const float* X            = (const float*)d_in[0];   // (65536, 512)
    const float* W            = (const float*)d_in[1];   // (512, 1024)
    const float* cluster_size = (const float*)d_in[2];   // (1024,)
    const float* embed_avg    = (const float*)d_in[3];   // (512, 1024)
    float* out = (float*)d_out;                          // (512, 1024)
    float* ws  = (float*)d_ws;

    float* esum   = ws + ESUM_OFF;
    float* counts = ws + COUNTS_OFF;
    float* w2     = ws + W2_OFF;
    float* cs     = ws + CS_OFF;

    // zero embed_sum + counts (contiguous at ws start)
    int zn = VEC * KCLUS + KCLUS;
    zero_ws_kernel<<<(zn + 255) / 256, 256, 0, stream>>>(ws, zn);

    w2_kernel<<<KCLUS / 256, 256, 0, stream>>>(W, w2);

    size_t smem = (size_t)ROWS_PER_BLOCK * XPAD * sizeof(float)
                + (size_t)ROWS_PER_BLOCK * sizeof(int);   // ~258.5 KB -> 1 WG/WGP
    assign_accum_kernel<<<BATCH / ROWS_PER_BLOCK, 128, smem, stream>>>(
        X, W, w2, esum, counts);

    finalize1_kernel<<<1, KCLUS, 0, stream>>>(cluster_size, counts, cs);
    finalize2_kernel<<<(VEC * KCLUS + 255) / 256, 256, 0, stream>>>(
        embed_avg, esum, cs, out);
}